// MobileNetYOLO_74448963109216
// MI455X (gfx1250) — compile-verified
//
#include <hip/hip_runtime.h>
#include <hip/hip_bf16.h>

// ---------------------------------------------------------------------------
// YOLO post-process for MI455X (gfx1250, wave32).
// Pipeline: [k1] per-anchor class max/argmax (bandwidth bound, ~129MB)
//           [k2] per-image radix-select top-2048 + LDS bitonic sort (O(N))
//           [k3] per-image greedy NMS with async-LDS staging + WMMA f32
//                AoS->SoA transpose, then final masked outputs.
// ---------------------------------------------------------------------------

#define BATCH   16
#define NANCH   25200
#define NCLASS  80
#define KTOP    2048
#define NTHREAD 1024
#define CONF_TH 0.25f
#define NMS_TH  0.45f

typedef __attribute__((ext_vector_type(2))) float v2f;
typedef __attribute__((ext_vector_type(8))) float v8f;

// ---------------------------------------------------------------------------
// Kernel 1: scores_all = obj * class_probs; max + first-wins argmax over 80
// classes; apply conf filter (zeroing) to the stored max score.
// One thread per anchor, 20 x float4 loads (320B contiguous per thread).
// ---------------------------------------------------------------------------
__global__ void k_score_max(const float* __restrict__ obj,
                            const float* __restrict__ probs,
                            float* __restrict__ maxScore,
                            int* __restrict__ labels,
                            int total) {
    int bn = blockIdx.x * blockDim.x + threadIdx.x;
    if (bn >= total) return;
    float o = obj[bn];
    const float4* p = (const float4*)probs + (size_t)bn * (NCLASS / 4);
    float mx = -1.0f;
    int lab = 0;
#pragma unroll
    for (int j = 0; j < NCLASS / 4; ++j) {
        float4 v = p[j];
        float s0 = o * v.x, s1 = o * v.y, s2 = o * v.z, s3 = o * v.w;
        int base = j * 4;
        if (s0 > mx) { mx = s0; lab = base;     }
        if (s1 > mx) { mx = s1; lab = base + 1; }
        if (s2 > mx) { mx = s2; lab = base + 2; }
        if (s3 > mx) { mx = s3; lab = base + 3; }
    }
    maxScore[bn] = (mx > CONF_TH) ? mx : 0.0f;   // conf filter (zeroed, not dropped)
    labels[bn]   = lab;
}

static __device__ __forceinline__ unsigned long long pack_key(unsigned bits, unsigned idx) {
    // Descending sort on this key == sort by (score desc, index asc):
    // scores are nonnegative floats so raw bits are order-preserving.
    return ((unsigned long long)bits << 32) | (unsigned)(~idx);
}

// ---------------------------------------------------------------------------
// Kernel 2: per-image (one 1024-thread workgroup each):
//   3-level radix select (bits 31:20, 19:8, 7:0) -> exact threshold T with
//   G = count(score > T) < K, then compact G ">" entries + (K-G) "==" entries,
//   bitonic-sort the 2048 packed keys descending in LDS, emit sorted
//   top scores/indices and gather boxes (AoS) + labels into workspace.
// ---------------------------------------------------------------------------
__global__ void k_select_sort(const float* __restrict__ maxScore,
                              const int* __restrict__ labels,
                              const float* __restrict__ boxes,
                              float* __restrict__ topScore,
                              int* __restrict__ topIdx,
                              int* __restrict__ topLabel,
                              float* __restrict__ topBox) {
    const int b   = blockIdx.x;
    const int tid = threadIdx.x;
    const int nth = blockDim.x;

    __shared__ unsigned int hist[4096];
    __shared__ unsigned long long keys[KTOP];
    __shared__ unsigned int sh_t, sh_need, sh_G, cG, cE;

    const float* s = maxScore + (size_t)b * NANCH;

    if (tid == 0) { sh_G = 0; sh_need = KTOP; }

    // ---- Level 1: bits [31:20] -------------------------------------------
    for (int i = tid; i < 4096; i += nth) hist[i] = 0;
    __syncthreads();
    for (int i = tid; i < NANCH; i += nth) {
        unsigned bits = __float_as_uint(s[i]);
        atomicAdd(&hist[bits >> 20], 1u);
    }
    __syncthreads();
    if (tid == 0) {
        unsigned need = sh_need, acc = 0;
        for (int v = 4095; v >= 0; --v) {
            unsigned h = hist[v];
            if (acc + h >= need) { sh_t = (unsigned)v; sh_G += acc; sh_need = need - acc; break; }
            acc += h;
        }
    }
    __syncthreads();
    const unsigned t1 = sh_t;

    // ---- Level 2: bits [19:8] within top-12 == t1 ------------------------
    for (int i = tid; i < 4096; i += nth) hist[i] = 0;
    __syncthreads();
    for (int i = tid; i < NANCH; i += nth) {
        unsigned bits = __float_as_uint(s[i]);
        if ((bits >> 20) == t1) atomicAdd(&hist[(bits >> 8) & 0xFFFu], 1u);
    }
    __syncthreads();
    if (tid == 0) {
        unsigned need = sh_need, acc = 0;
        for (int v = 4095; v >= 0; --v) {
            unsigned h = hist[v];
            if (acc + h >= need) { sh_t = (unsigned)v; sh_G += acc; sh_need = need - acc; break; }
            acc += h;
        }
    }
    __syncthreads();
    const unsigned t2 = sh_t;
    const unsigned pfx24 = (t1 << 12) | t2;

    // ---- Level 3: bits [7:0] within top-24 == pfx24 ----------------------
    for (int i = tid; i < 256; i += nth) hist[i] = 0;
    __syncthreads();
    for (int i = tid; i < NANCH; i += nth) {
        unsigned bits = __float_as_uint(s[i]);
        if ((bits >> 8) == pfx24) atomicAdd(&hist[bits & 0xFFu], 1u);
    }
    __syncthreads();
    if (tid == 0) {
        unsigned need = sh_need, acc = 0;
        for (int v = 255; v >= 0; --v) {
            unsigned h = hist[v];
            if (acc + h >= need) { sh_t = (unsigned)v; sh_G += acc; sh_need = need - acc; break; }
            acc += h;
        }
        cG = 0; cE = 0;
    }
    __syncthreads();
    const unsigned t3 = sh_t;
    const unsigned T  = (t1 << 20) | (t2 << 8) | t3;   // exact K-th largest bit pattern
    const unsigned G  = sh_G;                          // count(score > T), G < K

    // ---- Compaction -------------------------------------------------------
    for (int i = tid; i < NANCH; i += nth) {
        unsigned bits = __float_as_uint(s[i]);
        if (bits > T) {
            unsigned pos = atomicAdd(&cG, 1u);
            keys[pos] = pack_key(bits, (unsigned)i);
        } else if (bits == T) {
            unsigned pos = atomicAdd(&cE, 1u);
            if (pos < (unsigned)KTOP - G) keys[G + pos] = pack_key(bits, (unsigned)i);
        }
    }
    __syncthreads();

    // ---- Bitonic sort of 2048 keys, descending ---------------------------
    for (int k = 2; k <= KTOP; k <<= 1) {
        for (int j = k >> 1; j > 0; j >>= 1) {
            for (int i = tid; i < KTOP; i += nth) {
                int ixj = i ^ j;
                if (ixj > i) {
                    bool desc = ((i & k) == 0);
                    unsigned long long a = keys[i], c = keys[ixj];
                    bool sw = desc ? (a < c) : (a > c);
                    if (sw) { keys[i] = c; keys[ixj] = a; }
                }
            }
            __syncthreads();
        }
    }

    // ---- Emit sorted top-K + gathers -------------------------------------
    for (int i = tid; i < KTOP; i += nth) {
        unsigned long long key = keys[i];
        unsigned idx = ~((unsigned)key);
        float sc = __uint_as_float((unsigned)(key >> 32));
        int o = b * KTOP + i;
        topScore[o] = sc;
        topIdx[o]   = (int)idx;
        topLabel[o] = labels[(size_t)b * NANCH + idx];
        float4 bx = ((const float4*)boxes)[(size_t)b * NANCH + idx];
        ((float4*)topBox)[o] = bx;
    }
}

// ---------------------------------------------------------------------------
// Kernel 3: per-image greedy NMS.
//  - async-stage the 2048x4 f32 top boxes (AoS) into LDS (ASYNCcnt path)
//  - WMMA f32 16x16x4 with identity-A transposes each 16-box AoS tile into
//    SoA rows (x0/y0/x1/y1 striped across lanes) -- GLOBAL/DS_LOAD_TR* only
//    cover <=16-bit elements, so WMMA is the f32 transpose engine here.
//  - sequential i loop, j-parallel suppression, then masked outputs.
// ---------------------------------------------------------------------------
__global__ void k_nms(const float* __restrict__ topBox,
                      const float* __restrict__ topScore,
                      const int* __restrict__ topLabel,
                      float* __restrict__ out) {
    const int b   = blockIdx.x;
    const int tid = threadIdx.x;
    const int lane = tid & 31;

    __shared__ float stage[KTOP * 4];   // 32KB AoS staging (async load target)
    __shared__ float sx0[KTOP], sy0[KTOP], sx1[KTOP], sy1[KTOP];
    __shared__ float sarea[KTOP], sscore[KTOP];
    __shared__ unsigned char skeep[KTOP];

    // ---- Stage 1: async global -> LDS (b128 per thread-iteration) --------
    const float* src = topBox + (size_t)b * KTOP * 4;
#pragma unroll
    for (int k = 0; k < 2; ++k) {
        int i = tid + k * NTHREAD;                      // i in [0, 2048)
        unsigned ldsOff = (unsigned)(unsigned long long)&stage[i * 4];
        unsigned long long ga = (unsigned long long)(const void*)(src + i * 4);
        asm volatile("global_load_async_to_lds_b128 %0, %1, off"
                     :: "v"(ldsOff), "v"(ga) : "memory");
    }
    asm volatile("s_wait_asynccnt 0" ::: "memory");
    __syncthreads();

    // ---- Stage 2: WMMA AoS->SoA transpose, 16 boxes per wave-iteration ---
    // A = I (16x4): lanes 0-15 hold rows M=lane at K=0,1; lanes 16-31 rows
    // M=lane-16 at K=2,3.  B = coords x boxes (4x16): lane n<16 supplies
    // (x0,y0) of box n at K=0,1; lane n+16 supplies (x1,y1) at K=2,3.
    // D rows 0..3 (VGPRs 0..3, lanes 0-15) = x0/y0/x1/y1 across 16 boxes.
    {
        int wave = tid >> 5;
#pragma unroll
        for (int t = 0; t < 4; ++t) {
            int group  = wave * 4 + t;                  // 0..127 -> 2048 boxes
            int boxIdx = group * 16 + (lane & 15);
            int off    = (lane < 16) ? 0 : 2;
            v2f bb; bb.x = stage[boxIdx * 4 + off];
                    bb.y = stage[boxIdx * 4 + off + 1];
            v2f aa; aa.x = (lane == 0 || lane == 18) ? 1.0f : 0.0f;
                    aa.y = (lane == 1 || lane == 19) ? 1.0f : 0.0f;
            v8f c = {0.f, 0.f, 0.f, 0.f, 0.f, 0.f, 0.f, 0.f};
            v8f d = __builtin_amdgcn_wmma_f32_16x16x4_f32(
                        false, aa, false, bb, (short)0, c, false, false);
            if (lane < 16) {
                int idx = group * 16 + lane;
                float x0 = d[0], y0 = d[1], x1 = d[2], y1 = d[3];
                sx0[idx] = x0; sy0[idx] = y0; sx1[idx] = x1; sy1[idx] = y1;
                sarea[idx] = (x1 - x0) * (y1 - y0);
                float sc = topScore[b * KTOP + idx];
                sscore[idx] = sc;
                skeep[idx]  = (sc > 0.0f) ? 1 : 0;      // keep0 = score > 0
            }
        }
    }
    __syncthreads();

    // ---- Stage 3: greedy NMS (sequential over i, parallel over j) --------
    for (int i = 0; i < KTOP; ++i) {
        if (skeep[i]) {
            float x0i = sx0[i], y0i = sy0[i], x1i = sx1[i], y1i = sy1[i];
            float ai = sarea[i];
            for (int j = i + 1 + tid; j < KTOP; j += NTHREAD) {
                if (skeep[j]) {
                    float lx = fmaxf(x0i, sx0[j]);
                    float ly = fmaxf(y0i, sy0[j]);
                    float rx = fminf(x1i, sx1[j]);
                    float ry = fminf(y1i, sy1[j]);
                    float w = fmaxf(rx - lx, 0.0f);
                    float h = fmaxf(ry - ly, 0.0f);
                    float inter = w * h;
                    float iou = inter / (ai + sarea[j] - inter);
                    if (iou > NMS_TH) skeep[j] = 0;
                }
            }
        }
        __syncthreads();
    }

    // ---- Stage 4: masked outputs -----------------------------------------
    // d_out layout: boxes [B,K,4] | scores [B,K] | labels [B,K]
    float* outBoxes  = out;
    float* outScores = out + BATCH * KTOP * 4;
    float* outLabels = out + BATCH * KTOP * 5;
#pragma unroll
    for (int k = 0; k < 2; ++k) {
        int i = tid + k * NTHREAD;
        bool kp = skeep[i] != 0;
        float4 bx;
        bx.x = kp ? sx0[i] : 0.0f;
        bx.y = kp ? sy0[i] : 0.0f;
        bx.z = kp ? sx1[i] : 0.0f;
        bx.w = kp ? sy1[i] : 0.0f;
        ((float4*)outBoxes)[b * KTOP + i] = bx;
        outScores[b * KTOP + i] = kp ? sscore[i] : 0.0f;
        outLabels[b * KTOP + i] = kp ? (float)topLabel[b * KTOP + i] : 0.0f;
    }
}

// ---------------------------------------------------------------------------
extern "C" void kernel_launch(void* const* d_in, const int* in_sizes, int n_in,
                              void* d_out, int out_size, void* d_ws, size_t ws_size,
                              hipStream_t stream) {
    const float* boxes = (const float*)d_in[0];   // [B,N,4]
    const float* obj   = (const float*)d_in[1];   // [B,N]
    const float* probs = (const float*)d_in[2];   // [B,N,80]
    float* out = (float*)d_out;

    // Workspace layout (bytes)
    char* w = (char*)d_ws;
    size_t off = 0;
    float* maxScore = (float*)(w + off); off += (size_t)BATCH * NANCH * 4;
    int*   labels   = (int*)  (w + off); off += (size_t)BATCH * NANCH * 4;
    float* topScore = (float*)(w + off); off += (size_t)BATCH * KTOP * 4;
    int*   topIdx   = (int*)  (w + off); off += (size_t)BATCH * KTOP * 4;
    int*   topLabel = (int*)  (w + off); off += (size_t)BATCH * KTOP * 4;
    float* topBox   = (float*)(w + off); off += (size_t)BATCH * KTOP * 16;
    (void)ws_size; (void)n_in; (void)in_sizes; (void)out_size; (void)topIdx;

    const int total = BATCH * NANCH;
    k_score_max<<<(total + 255) / 256, 256, 0, stream>>>(obj, probs, maxScore, labels, total);
    k_select_sort<<<BATCH, NTHREAD, 0, stream>>>(maxScore, labels, boxes,
                                                 topScore, topIdx, topLabel, topBox);
    k_nms<<<BATCH, NTHREAD, 0, stream>>>(topBox, topScore, topLabel, out);
}